// MIAModel_1108101562889
// MI455X (gfx1250) — compile-verified
//
#include <hip/hip_runtime.h>
#include <math.h>

#define NN 8000
#define DD 256
#define PP 128

typedef __attribute__((ext_vector_type(2))) float v2f;
typedef __attribute__((ext_vector_type(4))) float v4f;
typedef __attribute__((ext_vector_type(8))) float v8f;

__device__ __forceinline__ float blockReduceSum256(float v, float* red) {
  const int t = threadIdx.x;
  red[t] = v;
  __syncthreads();
  for (int off = 128; off > 0; off >>= 1) {
    if (t < off) red[t] += red[t + off];
    __syncthreads();
  }
  float r = red[0];
  __syncthreads();
  return r;
}

// ---------------------------------------------------------------------------
// 1) deg -> dinv = rsqrt(row-sum(adj)).  Streams 256MB with NT hint.
// ---------------------------------------------------------------------------
__global__ void k_deg(const float* __restrict__ adj, float* __restrict__ dinv) {
  __shared__ float red[256];
  const int row = blockIdx.x;
  const v4f* a4 = reinterpret_cast<const v4f*>(adj + (size_t)row * NN);
  float s = 0.f;
  for (int i = threadIdx.x; i < NN / 4; i += 256) {
    v4f v = __builtin_nontemporal_load(&a4[i]);
    s += (v.x + v.y) + (v.z + v.w);
  }
  float tot = blockReduceSum256(s, red);
  if (threadIdx.x == 0) dinv[row] = rsqrtf(tot);
}

// ---------------------------------------------------------------------------
// 2) e1[n] = emb[n,:] . w_weight[0:D]
// ---------------------------------------------------------------------------
__global__ void k_e1(const float* __restrict__ emb, const float* __restrict__ wW,
                     float* __restrict__ e1) {
  __shared__ float red[256];
  const int row = blockIdx.x;
  float v = emb[(size_t)row * DD + threadIdx.x] * wW[threadIdx.x];
  float tot = blockReduceSum256(v, red);
  if (threadIdx.x == 0) e1[row] = tot;
}

// ---------------------------------------------------------------------------
// 3) proj[p,:] = pos_emb[p,:] @ w_att.T + b_att ; c2[p] = pos_emb[p,:].w2 ;
//    sdinv[p] = dinv[pos_idx[p]] (deduped: pos_idx is sorted, one-hot sets once)
// ---------------------------------------------------------------------------
__global__ void k_proj(const float* __restrict__ emb, const float* __restrict__ wAtt,
                       const float* __restrict__ bAtt, const float* __restrict__ wW,
                       const int* __restrict__ posIdx, const float* __restrict__ dinv,
                       float* __restrict__ proj, float* __restrict__ c2,
                       float* __restrict__ sdinv) {
  __shared__ float pe[DD];
  __shared__ float red[256];
  const int p = blockIdx.x;
  const int t = threadIdx.x;
  const int gi = posIdx[p];
  pe[t] = emb[(size_t)gi * DD + t];
  __syncthreads();
  float acc = 0.f;
  const float* wrow = wAtt + (size_t)t * DD;  // proj[p,d] = sum_k pe[k]*w_att[d,k]
  for (int k = 0; k < DD; ++k) acc = fmaf(pe[k], wrow[k], acc);
  proj[p * DD + t] = acc + bAtt[t];
  float tot = blockReduceSum256(pe[t] * wW[DD + t], red);
  if (t == 0) {
    c2[p] = tot;
    float sd = dinv[gi];
    if (p > 0 && posIdx[p - 1] == gi) sd = 0.f;  // duplicate index: one-hot counts once
    sdinv[p] = sd;
  }
}

// ---------------------------------------------------------------------------
// 4) Fused attention column kernel.  Per block: 16 columns (n), full P=128.
//    8 waves, each computes a 16x16 tile of proj @ emb.T with
//    v_wmma_f32_16x16x4_f32 (K=256 -> 64 WMMA per wave), then mask by
//    pos_adj, softmax over p (masked zeros participate, matching reference),
//    and fold straight into ctxw[n] (. c2), out1[n] (. sdinv), mutual_w[n].
// ---------------------------------------------------------------------------
__global__ void __launch_bounds__(256)
k_att(const float* __restrict__ emb, const float* __restrict__ adj,
      const int* __restrict__ posIdx, const float* __restrict__ proj,
      const float* __restrict__ c2, const float* __restrict__ sdinv,
      const float* __restrict__ dinv, const float* __restrict__ e1,
      float* __restrict__ out2) {
  __shared__ float sS[PP][16];   // masked scores tile
  __shared__ float r0[16][16];   // per-column reductions
  __shared__ float r1[16][16];
  __shared__ int   sPi[PP];
  __shared__ float sC2[PP];
  __shared__ float sSd[PP];

  const int tid = threadIdx.x;
  const int n0 = blockIdx.x * 16;
  if (tid < PP) { sPi[tid] = posIdx[tid]; sC2[tid] = c2[tid]; sSd[tid] = sdinv[tid]; }

  // ---- WMMA phase (EXEC all ones here) ----
  const int wv = tid >> 5;
  const int lane = tid & 31;
  const int halfk = (lane >> 4) << 1;            // lanes 16-31 hold K+2,K+3
  const int arow = (wv << 4) + (lane & 15);      // A: M = p row
  const int brow = n0 + (lane & 15);             // B: N = n row (emb row, since B=emb.T)

  const float* Abase = proj + (size_t)arow * DD + halfk;
  const float* Bbase = emb + (size_t)brow * DD + halfk;

  v8f c = {};
#pragma unroll 4
  for (int kk = 0; kk < DD; kk += 4) {
    v2f a = *reinterpret_cast<const v2f*>(Abase + kk);
    v2f b = *reinterpret_cast<const v2f*>(Bbase + kk);
    c = __builtin_amdgcn_wmma_f32_16x16x4_f32(false, a, false, b, (short)0, c,
                                              false, false);
  }

  // C/D layout: VGPR r -> (M=r, lanes 0-15) and (M=8+r, lanes 16-31)
  {
    const int pl = (wv << 4) + ((lane >> 4) << 3);
    const int nl = lane & 15;
#pragma unroll
    for (int r = 0; r < 8; ++r) sS[pl + r][nl] = c[r];
  }
  __syncthreads();

  const int col = tid & 15;   // local column
  const int sub = tid >> 4;   // 16 threads cooperate per column
  const int n = n0 + col;

  // pass 1: mask by binary pos_adj, column max, out1 partial
  float lmax = -3.4e38f, o1p = 0.f;
#pragma unroll
  for (int i = 0; i < 8; ++i) {
    const int p = (sub << 3) + i;
    const float m = adj[(size_t)sPi[p] * NN + n];  // 0.0 or 1.0
    const float v = sS[p][col] * m;
    sS[p][col] = v;
    lmax = fmaxf(lmax, v);
    o1p += sSd[p] * m;
  }
  r0[sub][col] = lmax;
  r1[sub][col] = o1p;
  __syncthreads();
  for (int off = 8; off > 0; off >>= 1) {
    if (sub < off) {
      r0[sub][col] = fmaxf(r0[sub][col], r0[sub + off][col]);
      r1[sub][col] += r1[sub + off][col];
    }
    __syncthreads();
  }
  const float cmax = r0[0][col];
  const float o1tot = r1[0][col];
  __syncthreads();

  // pass 2: softmax denominator + context-weight numerator
  float es = 0.f, ec = 0.f;
#pragma unroll
  for (int i = 0; i < 8; ++i) {
    const int p = (sub << 3) + i;
    const float e = __expf(sS[p][col] - cmax);
    es += e;
    ec += e * sC2[p];
  }
  r0[sub][col] = es;
  r1[sub][col] = ec;
  __syncthreads();
  for (int off = 8; off > 0; off >>= 1) {
    if (sub < off) {
      r0[sub][col] += r0[sub + off][col];
      r1[sub][col] += r1[sub + off][col];
    }
    __syncthreads();
  }
  if (sub == 0) {
    const float ctx = r1[0][col] / r0[0][col];               // context[n].w2
    const float mw = 1.f / (1.f + __expf(-(e1[n] + ctx)));   // sigmoid
    out2[n] = dinv[n] * o1tot * mw;                          // out1 * mutual_w
  }
}

// ---------------------------------------------------------------------------
// 5) out[j] = w_final * dinv[j] * (adj[j,:] . out2)   (adj symmetric)
// ---------------------------------------------------------------------------
__global__ void k_final(const float* __restrict__ adj, const float* __restrict__ out2,
                        const float* __restrict__ dinv, const float* __restrict__ wFin,
                        float* __restrict__ out) {
  __shared__ float red[256];
  const int row = blockIdx.x;
  const v4f* a4 = reinterpret_cast<const v4f*>(adj + (size_t)row * NN);
  const v4f* o4 = reinterpret_cast<const v4f*>(out2);
  float s = 0.f;
  for (int i = threadIdx.x; i < NN / 4; i += 256) {
    v4f a = __builtin_nontemporal_load(&a4[i]);
    v4f o = o4[i];
    s += a.x * o.x + a.y * o.y + a.z * o.z + a.w * o.w;
  }
  float tot = blockReduceSum256(s, red);
  if (threadIdx.x == 0) out[row] = tot * dinv[row] * wFin[0];
}

// ---------------------------------------------------------------------------
extern "C" void kernel_launch(void* const* d_in, const int* in_sizes, int n_in,
                              void* d_out, int out_size, void* d_ws, size_t ws_size,
                              hipStream_t stream) {
  (void)in_sizes; (void)n_in; (void)out_size; (void)ws_size;
  const float* adj    = (const float*)d_in[0];
  const float* emb    = (const float*)d_in[1];
  const float* wAtt   = (const float*)d_in[2];
  const float* bAtt   = (const float*)d_in[3];
  const float* wW     = (const float*)d_in[4];
  const float* wFin   = (const float*)d_in[5];
  const int*   posIdx = (const int*)d_in[6];
  float* out = (float*)d_out;

  float* ws    = (float*)d_ws;
  float* dinv  = ws;            // 8000 (pad to 8192)
  float* e1    = ws + 8192;     // 8000
  float* out2  = ws + 16384;    // 8000
  float* c2    = ws + 24576;    // 128
  float* sdinv = ws + 24704;    // 128
  float* proj  = ws + 24832;    // 128*256 = 32768

  k_deg  <<<NN,      256, 0, stream>>>(adj, dinv);
  k_e1   <<<NN,      256, 0, stream>>>(emb, wW, e1);
  k_proj <<<PP,      256, 0, stream>>>(emb, wAtt, bAtt, wW, posIdx, dinv, proj, c2, sdinv);
  k_att  <<<NN / 16, 256, 0, stream>>>(emb, adj, posIdx, proj, c2, sdinv, dinv, e1, out2);
  k_final<<<NN,      256, 0, stream>>>(adj, out2, dinv, wFin, out);
}